// MemoryEfficientCRF_40106404610170
// MI455X (gfx1250) — compile-verified
//
#include <hip/hip_runtime.h>
#include <stdint.h>

// CRF forward score on gfx1250 (scaled-probability form, single-barrier pipeline).
//   fv = m + log(p~/D); per step: q~ = p~ @ exp(Tr) (fp32 WMMA 16x16x4),
//   W = q~ * exp(emit) / D_prev,  Z = max_j W (deferred divisor for next step),
//   p~' = mask ? W : pstart,  D' = mask ? Z : 1,  m' = mask ? m+logZ : mStart (lazy).
// Block = 16 batches, 4 waves; wave w owns tags [16w,16w+16).
// Round-3 feedback: K-pair-interleaved P^T layout so every A fragment is a single
// aligned ds_load_b64 straight into a WMMA-legal VGPR pair (kills v_dual_mov repacking).

typedef float v8f __attribute__((ext_vector_type(8)));
typedef float v2f __attribute__((ext_vector_type(2)));

#define T_TAGS 64
#define SEQ    1024
#define BATCHN 512
#define PPAIR  17                 // pair-row stride in float2 units (bank-spread)
#define PROW   (2 * PPAIR)        // 34 floats per K-pair row
#define PELEMS (32 * PROW)        // 32 K-pair rows

// element (K, m) of P^T lives at (K>>1)*PROW + 2*m + (K&1)
#define PADDR(K, m) (((K) >> 1) * PROW + 2 * (m) + ((K) & 1))

// Fused 16-lane rotate-reduce: one VOP2-DPP instruction per step.
__device__ __forceinline__ float rowmax16(float v) {
    asm("v_max_num_f32_dpp %0, %0, %0 row_ror:8 row_mask:0xf bank_mask:0xf" : "+v"(v));
    asm("v_max_num_f32_dpp %0, %0, %0 row_ror:4 row_mask:0xf bank_mask:0xf" : "+v"(v));
    asm("v_max_num_f32_dpp %0, %0, %0 row_ror:2 row_mask:0xf bank_mask:0xf" : "+v"(v));
    asm("v_max_num_f32_dpp %0, %0, %0 row_ror:1 row_mask:0xf bank_mask:0xf" : "+v"(v));
    return v;
}
__device__ __forceinline__ float rowsum16(float v) {
    asm("v_add_f32_dpp %0, %0, %0 row_ror:8 row_mask:0xf bank_mask:0xf" : "+v"(v));
    asm("v_add_f32_dpp %0, %0, %0 row_ror:4 row_mask:0xf bank_mask:0xf" : "+v"(v));
    asm("v_add_f32_dpp %0, %0, %0 row_ror:2 row_mask:0xf bank_mask:0xf" : "+v"(v));
    asm("v_add_f32_dpp %0, %0, %0 row_ror:1 row_mask:0xf bank_mask:0xf" : "+v"(v));
    return v;
}

__global__ __launch_bounds__(128) void crf_fwd_wmma(
    const float* __restrict__ emissions,      // [B][S][T]
    const uint8_t* __restrict__ mask,         // [B][S] (bool, 1 byte)
    const float* __restrict__ transitions,    // [T][T]
    const float* __restrict__ start_t,        // [T]
    const float* __restrict__ end_t,          // [T]
    float* __restrict__ out)                  // [B]
{
    __shared__ __align__(16) float sP[2][PELEMS];   // P^T double buffer, K-pair interleaved
    __shared__ unsigned int sZ[3][16];              // per-batch max(W) bits, triple-buffered
    __shared__ float        sMask[2][16];           // staged mask column (float)
    __shared__ float        sM[16];                 // running log-scale m (lazily updated)
    __shared__ float        sSum[16];               // final logsumexp accumulator

    const int tid   = threadIdx.x;
    const int wv    = tid >> 5;          // wave 0..3 -> tag tile
    const int lane  = tid & 31;
    const int l16   = lane & 15;
    const int half8 = (lane >> 4) << 3;  // 0 (lanes 0-15) or 8 (lanes 16-31)
    const int tag   = (wv << 4) | l16;   // this lane's tag column (N)
    const int b0    = blockIdx.x << 4;   // 16 batches per block

    if (tid < 16) {
        sZ[0][tid] = 0u; sZ[1][tid] = 0u; sZ[2][tid] = 0u;
        sSum[tid] = 0.f; sM[tid] = 0.f;
        sMask[0][tid] = 1.f;             // virtual mask for step 0 (unconditional)
    }

    // ---- resident B fragments: B[K][N] = exp(Tr[K][tag]), 16 K-chunks of 4 ----
    // f32 4x16 B layout: VGPR0: K = 4k+0 (lanes 0-15) / 4k+2 (lanes 16-31); VGPR1: +1
    const int koff = (lane < 16) ? 0 : 2;
    v2f Bfrag[16];
#pragma unroll
    for (int k = 0; k < 16; ++k) {
        Bfrag[k].x = __expf(transitions[(4 * k + koff)     * T_TAGS + tag]);
        Bfrag[k].y = __expf(transitions[(4 * k + koff + 1) * T_TAGS + tag]);
    }

    const float sv = start_t[tag];
    const float ev = end_t[tag];
    float mStart = -INFINITY;
    for (int j = 0; j < T_TAGS; ++j) mStart = fmaxf(mStart, start_t[j]);
    const float pstart = __expf(sv - mStart);  // start repr: p=pstart, D=1, m=mStart

    // A-fragment base: chunk k reads K-pair row (2k + kof1), batch column l16
    const int kof1 = (lane < 16) ? 0 : 1;     // K=4k+{0,1} vs K=4k+{2,3}
    const int aBase = kof1 * PROW + 2 * l16;  // + k*2*PROW per chunk (imm offsets)
    const int stAddr = PADDR(tag, half8);     // + 2*r per accumulator row

    // emissions: per (batch, tag) element advances by T_TAGS per step (contiguous)
    const float* ep = emissions + (size_t)(b0 + half8) * SEQ * T_TAGS + tag;
    float eraw[8];
#pragma unroll
    for (int r = 0; r < 8; ++r) eraw[r] = ep[(size_t)r * SEQ * T_TAGS]; // t = 0

    __syncthreads();  // LDS init visible

    float p[8];   // p~ slice (batch r+half8, tag), un-normalized (divisor deferred)
    float mi[8];  // mask column of the step that produced p~

    // ================= step 0: p~(0) = exp(start + emit0), D(0) = Z(0) =================
    {
#pragma unroll
        for (int r = 0; r < 8; ++r) {
            float w = __expf(sv + eraw[r]);
            p[r] = w;
            mi[r] = 1.f;
            float v = rowmax16(w);
            if (l16 == 0) atomicMax(&sZ[0][r + half8], __float_as_uint(v));
            sP[0][stAddr + 2 * r] = w;
        }
        // prefetch emissions + mask for t = 1
#pragma unroll
        for (int r = 0; r < 8; ++r) eraw[r] = ep[(size_t)r * SEQ * T_TAGS + T_TAGS];
        if (tid < 16) sMask[1][tid] = (float)mask[(size_t)(b0 + tid) * SEQ + 1];
        __syncthreads();  // Z(0), P(0), mask(1) complete
    }

    // ================= scan: t = 1 .. SEQ-1 (ONE barrier per step) =================
    int zprev = 0, zcur = 1, znext = 2;
    for (int t = 1; t < SEQ; ++t) {
        // deferred divisor from previous step: D = mi_prev ? Z(t-1) : 1
        float er[8], mt[8];
#pragma unroll
        for (int r = 0; r < 8; ++r) {
            float Zf = __uint_as_float(sZ[zprev][r + half8]);
            float rd = mi[r] * (__builtin_amdgcn_rcpf(Zf) - 1.f) + 1.f;
            er[r] = __expf(eraw[r]) * rd;          // e_t / D(t-1)
            mt[r] = sMask[t & 1][r + half8];       // mask column t
        }

        // prefetch t+1 early so HBM latency hides under the matmul
        const int tn = (t + 1 < SEQ) ? (t + 1) : (SEQ - 1);
        const float* epn = ep + (size_t)tn * T_TAGS;
#pragma unroll
        for (int r = 0; r < 8; ++r) eraw[r] = epn[(size_t)r * SEQ * T_TAGS];

        // A fragments from P~(t-1): one aligned ds_load_b64 each (no repacking)
        const float* Pin = sP[(t + 1) & 1];        // == (t-1)&1
        float*       Pout = sP[t & 1];
        v2f A[16];
#pragma unroll
        for (int k = 0; k < 16; ++k)
            A[k] = *(const v2f*)(Pin + aBase + k * 2 * PROW);

        // q~ = P~(16x64) @ expT(64x16): 16 WMMAs in 4 independent chains (depth 4)
        v8f a0 = {0.f,0.f,0.f,0.f,0.f,0.f,0.f,0.f};
        v8f a1 = {0.f,0.f,0.f,0.f,0.f,0.f,0.f,0.f};
        v8f a2 = {0.f,0.f,0.f,0.f,0.f,0.f,0.f,0.f};
        v8f a3 = {0.f,0.f,0.f,0.f,0.f,0.f,0.f,0.f};
#pragma unroll
        for (int k = 0; k < 4; ++k) {
            a0 = __builtin_amdgcn_wmma_f32_16x16x4_f32(false, A[4*k+0], false, Bfrag[4*k+0], (short)0, a0, false, false);
            a1 = __builtin_amdgcn_wmma_f32_16x16x4_f32(false, A[4*k+1], false, Bfrag[4*k+1], (short)0, a1, false, false);
            a2 = __builtin_amdgcn_wmma_f32_16x16x4_f32(false, A[4*k+2], false, Bfrag[4*k+2], (short)0, a2, false, false);
            a3 = __builtin_amdgcn_wmma_f32_16x16x4_f32(false, A[4*k+3], false, Bfrag[4*k+3], (short)0, a3, false, false);
        }

#pragma unroll
        for (int r = 0; r < 8; ++r) {
            float w = ((a0[r] + a1[r]) + (a2[r] + a3[r])) * er[r];   // W(t)
            // per-batch max over 64 tags -> Z(t) (cross-wave via LDS uint max; W > 0)
            float v = rowmax16(w);
            if (l16 == 0) atomicMax(&sZ[zcur][r + half8], __float_as_uint(v));
            // mask blend + store p~(t) (un-normalized; divisor applied next step)
            float pb = fmaf(mt[r], w - pstart, pstart);
            Pout[stAddr + 2 * r] = pb;
            p[r]  = pb;
            mi[r] = mt[r];
        }

        // wave0 housekeeping (reads of col t-1 precede the staging write; same-lane
        // DS ordering keeps this safe within the single barrier interval)
        if (tid < 16) {
            float Zp = __uint_as_float(sZ[zprev][tid]);
            float mp = sMask[(t + 1) & 1][tid];    // mask column t-1 (same parity buf)
            sM[tid] = mp * (sM[tid] + __logf(Zp)) + (1.f - mp) * mStart;  // lazy m(t-1)
            sMask[tn & 1][tid] = (float)mask[(size_t)(b0 + tid) * SEQ + tn]; // stage col t+1
            sZ[znext][tid] = 0u;                   // re-arm (held Z(t-2), last read t-1)
        }

        __syncthreads();  // Z(t), P~(t), mask(t+1), m(t-1) all complete

        int zt = zprev; zprev = zcur; zcur = znext; znext = zt;
    }

    // ============ finalize: out = m(S-1) + log sum_j (p~/D) * exp(end_j * mask_last) ============
#pragma unroll
    for (int r = 0; r < 8; ++r) {
        float Zf = __uint_as_float(sZ[zprev][r + half8]);   // Z(S-1)
        float rd = mi[r] * (__builtin_amdgcn_rcpf(Zf) - 1.f) + 1.f;
        float c = rowsum16(p[r] * rd * __expf(ev * mi[r])); // mi holds mask[:, S-1]
        if (l16 == 0) atomicAdd(&sSum[r + half8], c);
    }
    __syncthreads();
    if (tid < 16) {
        float Zf = __uint_as_float(sZ[zprev][tid]);
        float ml = sMask[(SEQ - 1) & 1][tid];               // mask[:, S-1]
        float mf = ml * (sM[tid] + __logf(Zf)) + (1.f - ml) * mStart;
        out[b0 + tid] = mf + __logf(sSum[tid]);
    }
}

extern "C" void kernel_launch(void* const* d_in, const int* in_sizes, int n_in,
                              void* d_out, int out_size, void* d_ws, size_t ws_size,
                              hipStream_t stream) {
    (void)in_sizes; (void)n_in; (void)out_size; (void)d_ws; (void)ws_size;
    const float*   emissions   = (const float*)d_in[0];
    const uint8_t* mask        = (const uint8_t*)d_in[1];
    const float*   transitions = (const float*)d_in[2];
    const float*   start_t     = (const float*)d_in[3];
    const float*   end_t       = (const float*)d_in[4];
    float*         out         = (float*)d_out;

    dim3 grid(BATCHN / 16);   // 32 blocks x 16 batches
    dim3 block(128);          // 4 waves: one 16-tag WMMA column tile each
    crf_fwd_wmma<<<grid, block, 0, stream>>>(emissions, mask, transitions,
                                             start_t, end_t, out);
}